// DSP2Net_266287972641
// MI455X (gfx1250) — compile-verified
//
#include <hip/hip_runtime.h>
#include <hip/hip_bf16.h>

// ---------------- problem constants ----------------
#define BB    32
#define C3    32
#define CI    64
#define DD    32
#define HH    32
#define WW    32
#define DIM   64
#define NTOK  1025   // 1 + 32*32
#define EPS   1e-5f

typedef __attribute__((ext_vector_type(2))) float v2f;
typedef __attribute__((ext_vector_type(8))) float v8f;

// D = A(16x4) * B(4x16) + C, fp32 WMMA (CDNA5 V_WMMA_F32_16X16X4_F32)
static __device__ inline v8f wmma_f32_4(v2f a, v2f b, v8f c) {
    return __builtin_amdgcn_wmma_f32_16x16x4_f32(
        /*neg_a=*/false, a, /*neg_b=*/false, b,
        /*c_mod=*/(short)0, c, /*reuse_a=*/false, /*reuse_b=*/false);
}

// weights matrix for implicit-GEMM conv3d: K = 54 taps (27 dil-1 + 27 dil-2),
// padded to 56; branch average folded in as 0.5.
static __device__ inline float wmat3d(const float* __restrict__ w1,
                                      const float* __restrict__ w2, int k, int n) {
    if (k >= 54) return 0.f;
    return 0.5f * ((k < 27) ? w1[n * 27 + k] : w2[n * 27 + (k - 27)]);
}

// per-lane LDS offset (relative to (z+2)*100) for conv tap k at w-col n15.
// k >= 54 pairs with a zero B-row, so any valid slab address works -> 0.
static __device__ inline int tap_off(int k, int n15) {
    if (k >= 54) return 0;
    int t  = (k < 27) ? k : (k - 27);
    int dz = t / 9 - 1, dy = (t % 9) / 3 - 1, dx = t % 3 - 1;
    if (k >= 27) { dz <<= 1; dy <<= 1; dx <<= 1; }
    return dz * 100 + (dy + 2) * 20 + (n15 + 2 + dx);
}

// stage x slab for tile (b,h,w0): z -2..33, h-2..h+2, w0-2..w0+17 (zero padded)
static __device__ inline void load_slab(float* __restrict__ slab,
                                        const float* __restrict__ x,
                                        int b, int h, int w0, int lane) {
    for (int idx = lane; idx < 3600; idx += 32) {
        int zz = idx / 100, rem = idx % 100;
        int r = rem / 20, c = rem % 20;
        int gz = zz - 2, gh = h - 2 + r, gw = w0 - 2 + c;
        float v = 0.f;
        if ((unsigned)gz < 32u && (unsigned)gh < 32u && (unsigned)gw < 32u)
            v = x[((b * DD + gz) * HH + gh) * WW + gw];
        slab[idx] = v;
    }
    __syncthreads();
}

// =====================================================================
// K1: conv3d (both dilated branches, averaged) -> per-channel partial
//     sum / sumsq per tile (deterministic BN train stats, pass 1).
// one wave per tile = (b, h, w0 in {0,16}); per z-slice: batch 28 DS
// loads into registers, then run the 28-WMMA chain (loads pipeline).
// =====================================================================
__global__ void k_conv_stats(const float* __restrict__ x,
                             const float* __restrict__ w1,
                             const float* __restrict__ w2,
                             float* __restrict__ part) {
    __shared__ float slab[3600];
    int tile = blockIdx.x;                 // 2048
    int lane = threadIdx.x;                // 32
    int w0 = (tile & 1) * 16;
    int h  = (tile >> 1) & 31;
    int b  = tile >> 6;
    int n15 = lane & 15, hi = lane >> 4;

    load_slab(slab, x, b, h, w0, lane);

    v2f bw[14][2];
    int d0[14], d1[14];
#pragma unroll
    for (int s = 0; s < 14; ++s) {
        int r0 = 4 * s + 2 * hi;
        d0[s] = tap_off(r0, n15);
        d1[s] = tap_off(r0 + 1, n15);
#pragma unroll
        for (int nt = 0; nt < 2; ++nt) {
            int n = nt * 16 + n15;
            bw[s][nt][0] = wmat3d(w1, w2, r0, n);
            bw[s][nt][1] = wmat3d(w1, w2, r0 + 1, n);
        }
    }
    float s0 = 0.f, q0 = 0.f, s1 = 0.f, q1 = 0.f;
    for (int z = 0; z < 32; ++z) {
        int base = (z + 2) * 100;
        v2f a[14];
#pragma unroll
        for (int s = 0; s < 14; ++s) {     // batch all A loads first
            a[s][0] = slab[base + d0[s]];
            a[s][1] = slab[base + d1[s]];
        }
        v8f c0 = {}; v8f c1 = {};
#pragma unroll
        for (int s = 0; s < 14; ++s) {     // then the WMMA chain
            c0 = wmma_f32_4(a[s], bw[s][0], c0);
            c1 = wmma_f32_4(a[s], bw[s][1], c1);
        }
#pragma unroll
        for (int i = 0; i < 8; ++i) {
            float e0 = c0[i]; s0 += e0; q0 += e0 * e0;
            float e1 = c1[i]; s1 += e1; q1 += e1 * e1;
        }
    }
    s0 += __shfl_down(s0, 16); q0 += __shfl_down(q0, 16);
    s1 += __shfl_down(s1, 16); q1 += __shfl_down(q1, 16);
    if (lane < 16) {
        float* p = part + (size_t)tile * 128;
        p[lane]      = s0;  p[16 + lane]  = s1;
        p[64 + lane] = q0;  p[80 + lane]  = q1;
    }
}

// finalize BN3 stats: mean / inv-std per channel (count = B*D*H*W)
__global__ void k_fin3(const float* __restrict__ part, float* __restrict__ fin) {
    int c = threadIdx.x;                   // 32
    float S = 0.f, Q = 0.f;
    for (int t = 0; t < 2048; ++t) {
        S += part[(size_t)t * 128 + c];
        Q += part[(size_t)t * 128 + 64 + c];
    }
    const float cnt = 1048576.f;
    float m = S / cnt;
    float v = Q / cnt - m * m;
    fin[c] = m; fin[32 + c] = rsqrtf(v + EPS);
}

// =====================================================================
// K2: conv3d recompute -> BN(affine) -> ReLU -> mean over z  => y2
// =====================================================================
__global__ void k_conv_apply(const float* __restrict__ x,
                             const float* __restrict__ w1,
                             const float* __restrict__ w2,
                             const float* __restrict__ bn_g,
                             const float* __restrict__ bn_b,
                             const float* __restrict__ fin,
                             float* __restrict__ y2) {
    __shared__ float slab[3600];
    int tile = blockIdx.x;                 // 2048
    int lane = threadIdx.x;
    int w0 = (tile & 1) * 16;
    int h  = (tile >> 1) & 31;
    int b  = tile >> 6;
    int n15 = lane & 15, hi = lane >> 4;

    load_slab(slab, x, b, h, w0, lane);

    v2f bw[14][2];
    int d0[14], d1[14];
#pragma unroll
    for (int s = 0; s < 14; ++s) {
        int r0 = 4 * s + 2 * hi;
        d0[s] = tap_off(r0, n15);
        d1[s] = tap_off(r0 + 1, n15);
#pragma unroll
        for (int nt = 0; nt < 2; ++nt) {
            int n = nt * 16 + n15;
            bw[s][nt][0] = wmat3d(w1, w2, r0, n);
            bw[s][nt][1] = wmat3d(w1, w2, r0 + 1, n);
        }
    }
    int ch0 = n15, ch1 = 16 + n15;
    float sc0 = fin[32 + ch0] * bn_g[ch0];
    float of0 = bn_b[ch0] - fin[ch0] * sc0;
    float sc1 = fin[32 + ch1] * bn_g[ch1];
    float of1 = bn_b[ch1] - fin[ch1] * sc1;

    float acc0[8] = {0, 0, 0, 0, 0, 0, 0, 0};
    float acc1[8] = {0, 0, 0, 0, 0, 0, 0, 0};
    for (int z = 0; z < 32; ++z) {
        int base = (z + 2) * 100;
        v2f a[14];
#pragma unroll
        for (int s = 0; s < 14; ++s) {     // batch all A loads first
            a[s][0] = slab[base + d0[s]];
            a[s][1] = slab[base + d1[s]];
        }
        v8f c0 = {}; v8f c1 = {};
#pragma unroll
        for (int s = 0; s < 14; ++s) {     // then the WMMA chain
            c0 = wmma_f32_4(a[s], bw[s][0], c0);
            c1 = wmma_f32_4(a[s], bw[s][1], c1);
        }
#pragma unroll
        for (int i = 0; i < 8; ++i) {
            acc0[i] += fmaxf(0.f, c0[i] * sc0 + of0);
            acc1[i] += fmaxf(0.f, c1[i] * sc1 + of1);
        }
    }
#pragma unroll
    for (int i = 0; i < 8; ++i) {
        int M = i + (hi ? 8 : 0);
        int w = w0 + M;
        y2[((b * C3 + ch0) * HH + h) * WW + w] = acc0[i] * (1.f / 32.f);
        y2[((b * C3 + ch1) * HH + h) * WW + w] = acc1[i] * (1.f / 32.f);
    }
}

// =====================================================================
// K3: involution front-end.  A_pw = depthwise3x3(y2), A_red = y2 (from LDS);
//     WMMA x (w_pw, w_red) -> pw[B,64,H,W], red[B,64,H,W] + red BN partials
// =====================================================================
__global__ void k_inv_proj(const float* __restrict__ y2,
                           const float* __restrict__ w_dw,
                           const float* __restrict__ w_pw,
                           const float* __restrict__ w_red,
                           float* __restrict__ pw,
                           float* __restrict__ red,
                           float* __restrict__ part2) {
    __shared__ float y2s[32 * 3 * 18];     // [c][h-1..h+1][w0-1..w0+16]
    __shared__ float wdws[C3 * 9];
    int tile = blockIdx.x;                 // 2048
    int lane = threadIdx.x;
    int w0 = (tile & 1) * 16;
    int h  = (tile >> 1) & 31;
    int b  = tile >> 6;
    int n15 = lane & 15, hi = lane >> 4;

    for (int idx = lane; idx < 32 * 54; idx += 32) {
        int c = idx / 54, rem = idx % 54;
        int r = rem / 18, cc = rem % 18;
        int gh = h - 1 + r, gw = w0 - 1 + cc;
        float v = 0.f;
        if ((unsigned)gh < 32u && (unsigned)gw < 32u)
            v = y2[((b * C3 + c) * HH + gh) * WW + gw];
        y2s[idx] = v;
    }
    for (int idx = lane; idx < C3 * 9; idx += 32) wdws[idx] = w_dw[idx];
    __syncthreads();

    v2f Ar[8], Ap[8];
#pragma unroll
    for (int s = 0; s < 8; ++s) {
        int c0 = 4 * s + 2 * hi;
#pragma unroll
        for (int u = 0; u < 2; ++u) {
            int c = c0 + u;
            const float* cs = y2s + c * 54;
            Ar[s][u] = cs[18 + n15 + 1];
            float dv = 0.f;
#pragma unroll
            for (int t = 0; t < 9; ++t)
                dv += cs[(t / 3) * 18 + (n15 + t % 3)] * wdws[c * 9 + t];
            Ap[s][u] = dv;
        }
    }
#pragma unroll
    for (int nt = 0; nt < 4; ++nt) {
        int n = nt * 16 + n15;
        v8f cp = {}; v8f cr = {};
#pragma unroll
        for (int s = 0; s < 8; ++s) {
            int r0 = 4 * s + 2 * hi;
            v2f bp, br;
            bp[0] = w_pw[n * 32 + r0];  bp[1] = w_pw[n * 32 + r0 + 1];
            br[0] = w_red[n * 32 + r0]; br[1] = w_red[n * 32 + r0 + 1];
            cp = wmma_f32_4(Ap[s], bp, cp);
            cr = wmma_f32_4(Ar[s], br, cr);
        }
        float S = 0.f, Q = 0.f;
#pragma unroll
        for (int i = 0; i < 8; ++i) {
            int M = i + (hi ? 8 : 0);
            int w = w0 + M;
            pw[((b * CI + n) * HH + h) * WW + w] = cp[i];
            float e = cr[i];
            red[((b * CI + n) * HH + h) * WW + w] = e;
            S += e; Q += e * e;
        }
        S += __shfl_down(S, 16); Q += __shfl_down(Q, 16);
        if (lane < 16) {
            part2[(size_t)tile * 128 + n]      = S;
            part2[(size_t)tile * 128 + 64 + n] = Q;
        }
    }
}

// finalize BN2 stats (count = B*H*W)
__global__ void k_fin2(const float* __restrict__ part, float* __restrict__ fin) {
    int c = threadIdx.x;                   // 64
    float S = 0.f, Q = 0.f;
    for (int t = 0; t < 2048; ++t) {
        S += part[(size_t)t * 128 + c];
        Q += part[(size_t)t * 128 + 64 + c];
    }
    const float cnt = 32768.f;
    float m = S / cnt;
    float v = Q / cnt - m * m;
    fin[c] = m; fin[64 + c] = rsqrtf(v + EPS);
}

// =====================================================================
// K4: kernel generation (BN+ReLU+span) and involution gather -> tokens
// one block per (b,h); thread = w.  cls token written by h==0 blocks.
// =====================================================================
__global__ void k_involution_tok(const float* __restrict__ pw,
                                 const float* __restrict__ red,
                                 const float* __restrict__ fin2,
                                 const float* __restrict__ bn2_g,
                                 const float* __restrict__ bn2_b,
                                 const float* __restrict__ w_span,
                                 const float* __restrict__ cls,
                                 float* __restrict__ tok) {
    __shared__ float pw_s[64 * 3 * 34];    // [c][3 rows][w -1..32]
    __shared__ float span_s[9 * 64];
    int bid = blockIdx.x;                  // 1024
    int b = bid >> 5, h = bid & 31;
    int tid = threadIdx.x;                 // 32

    for (int idx = tid; idx < 64 * 3 * 34; idx += 32) {
        int c = idx / 102, rem = idx % 102;
        int r = rem / 34, ww = rem % 34 - 1;
        int hh = h - 1 + r;
        float v = 0.f;
        if ((unsigned)hh < 32u && (unsigned)ww < 32u)
            v = pw[((b * CI + c) * HH + hh) * WW + ww];
        pw_s[idx] = v;
    }
    for (int idx = tid; idx < 576; idx += 32) span_s[idx] = w_span[idx];
    __syncthreads();

    int w = tid;
    float kern[9] = {0, 0, 0, 0, 0, 0, 0, 0, 0};
    for (int o = 0; o < 64; ++o) {
        float sc = fin2[64 + o] * bn2_g[o];
        float of = bn2_b[o] - fin2[o] * sc;
        float r  = red[((b * CI + o) * HH + h) * WW + w];
        float rn = fmaxf(0.f, r * sc + of);
#pragma unroll
        for (int t = 0; t < 9; ++t) kern[t] += span_s[t * 64 + o] * rn;
    }
    float* trow = tok + (size_t)(b * NTOK + 1 + h * WW + w) * DIM;
    for (int o = 0; o < 64; ++o) {
        const float* ps = pw_s + o * 102;
        float s = 0.f;
#pragma unroll
        for (int t = 0; t < 9; ++t)
            s += ps[(t / 3) * 34 + (w + t % 3)] * kern[t];
        trow[o] = s;
    }
    if (h == 0) {
        for (int o = tid; o < 64; o += 32)
            tok[(size_t)b * NTOK * DIM + o] = cls[o];
    }
}

// =====================================================================
// K5: K,V projections  (tok[32800,64] @ wk/wv[64,64]) via fp32 WMMA
// one wave per 16 token rows; A fragments loaded as b64
// =====================================================================
__global__ void k_kv_proj(const float* __restrict__ tok,
                          const float* __restrict__ wk,
                          const float* __restrict__ wv,
                          float* __restrict__ kbuf,
                          float* __restrict__ vbuf) {
    int m0 = blockIdx.x * 16;              // 2050 blocks, 32800 rows total
    int lane = threadIdx.x;
    int n15 = lane & 15, hi = lane >> 4;
    int row = m0 + n15;

    v2f A[16];
#pragma unroll
    for (int s = 0; s < 16; ++s) {
        int k0 = 4 * s + 2 * hi;
        A[s] = *(const v2f*)(tok + (size_t)row * DIM + k0);
    }
#pragma unroll
    for (int nt = 0; nt < 4; ++nt) {
        int n = nt * 16 + n15;
        v8f ck = {}; v8f cv = {};
#pragma unroll
        for (int s = 0; s < 16; ++s) {
            int r0 = 4 * s + 2 * hi;
            v2f bk, bv;
            bk[0] = wk[r0 * DIM + n]; bk[1] = wk[(r0 + 1) * DIM + n];
            bv[0] = wv[r0 * DIM + n]; bv[1] = wv[(r0 + 1) * DIM + n];
            ck = wmma_f32_4(A[s], bk, ck);
            cv = wmma_f32_4(A[s], bv, cv);
        }
#pragma unroll
        for (int i = 0; i < 8; ++i) {
            int r = m0 + i + (hi ? 8 : 0);
            kbuf[(size_t)r * DIM + n] = ck[i];
            vbuf[(size_t)r * DIM + n] = cv[i];
        }
    }
}

// =====================================================================
// K6: cross-attention (1 query = cls), softmax over 1025, out-proj,
//     exact-GELU FFN, residual.  one block per batch element.
// =====================================================================
__global__ void k_attn_ffn(const float* __restrict__ tok,
                           const float* __restrict__ kbuf,
                           const float* __restrict__ vbuf,
                           const float* __restrict__ wq,
                           const float* __restrict__ wo,
                           const float* __restrict__ bo,
                           const float* __restrict__ fw1,
                           const float* __restrict__ fb1,
                           const float* __restrict__ fw2,
                           const float* __restrict__ fb2,
                           float* __restrict__ out) {
    __shared__ float q_s[64];
    __shared__ float dots_s[8 * NTOK];
    __shared__ float opart[4 * 64];
    __shared__ float o_s[64];
    __shared__ float os_s[64];
    __shared__ float h1_s[512];
    int b = blockIdx.x;                    // 32
    int tid = threadIdx.x;                 // 256
    const float* crow = tok + (size_t)b * NTOK * DIM;   // cls token row

    if (tid < 64) {
        float s = 0.f;
        for (int i = 0; i < 64; ++i) s += crow[i] * wq[i * DIM + tid];
        q_s[tid] = s;
    }
    __syncthreads();

    const float scale = 0.35355339059327373f;   // 8^-0.5
    for (int idx = tid; idx < 8 * NTOK; idx += 256) {
        int j = idx >> 3, hh = idx & 7;
        const float* kr = kbuf + ((size_t)b * NTOK + j) * DIM + hh * 8;
        const float* qh = q_s + hh * 8;
        float d = 0.f;
#pragma unroll
        for (int t = 0; t < 8; ++t) d += qh[t] * kr[t];
        dots_s[hh * NTOK + j] = d * scale;
    }
    __syncthreads();

    {   // softmax: one wave per head
        int hh = tid >> 5, lane = tid & 31;
        float* dr = dots_s + hh * NTOK;
        float m = -1e30f;
        for (int j = lane; j < NTOK; j += 32) m = fmaxf(m, dr[j]);
        for (int o = 16; o > 0; o >>= 1) m = fmaxf(m, __shfl_xor(m, o));
        float s = 0.f;
        for (int j = lane; j < NTOK; j += 32) {
            float e = expf(dr[j] - m); dr[j] = e; s += e;
        }
        for (int o = 16; o > 0; o >>= 1) s += __shfl_xor(s, o);
        float inv = 1.f / s;
        for (int j = lane; j < NTOK; j += 32) dr[j] *= inv;
    }
    __syncthreads();

    {   // o[hh][d] = sum_j attn * v
        int oi = tid & 63, p = tid >> 6;
        int hh = oi >> 3, d = oi & 7;
        float s = 0.f;
        for (int j = p; j < NTOK; j += 4)
            s += dots_s[hh * NTOK + j] *
                 vbuf[((size_t)b * NTOK + j) * DIM + hh * 8 + d];
        opart[p * 64 + oi] = s;
    }
    __syncthreads();
    if (tid < 64)
        o_s[tid] = opart[tid] + opart[64 + tid] + opart[128 + tid] + opart[192 + tid];
    __syncthreads();

    if (tid < 64) {                        // out projection + bias
        float s = bo[tid];
        for (int i = 0; i < 64; ++i) s += o_s[i] * wo[i * DIM + tid];
        os_s[tid] = s;
    }
    __syncthreads();

    for (int m = tid; m < 512; m += 256) { // FFN up + exact GELU
        float s = fb1[m];
        for (int i = 0; i < 64; ++i) s += os_s[i] * fw1[i * 512 + m];
        h1_s[m] = 0.5f * s * (1.f + erff(s * 0.70710678118654752f));
    }
    __syncthreads();

    if (tid < 64) {                        // FFN down + residual
        float s = os_s[tid] + fb2[tid];
        for (int m = 0; m < 512; ++m) s += h1_s[m] * fw2[m * DIM + tid];
        out[b * DIM + tid] = s;
    }
}

// =====================================================================
extern "C" void kernel_launch(void* const* d_in, const int* in_sizes, int n_in,
                              void* d_out, int out_size, void* d_ws, size_t ws_size,
                              hipStream_t stream) {
    (void)in_sizes; (void)n_in; (void)out_size; (void)ws_size;
    const float* x     = (const float*)d_in[0];
    const float* w3d1  = (const float*)d_in[1];
    const float* w3d2  = (const float*)d_in[2];
    const float* bn3g  = (const float*)d_in[3];
    const float* bn3b  = (const float*)d_in[4];
    const float* w_dw  = (const float*)d_in[5];
    const float* w_pw  = (const float*)d_in[6];
    const float* w_red = (const float*)d_in[7];
    const float* bn2g  = (const float*)d_in[8];
    const float* bn2b  = (const float*)d_in[9];
    const float* w_spn = (const float*)d_in[10];
    const float* cls   = (const float*)d_in[11];
    const float* wq    = (const float*)d_in[12];
    const float* wk    = (const float*)d_in[13];
    const float* wv    = (const float*)d_in[14];
    const float* wo    = (const float*)d_in[15];
    const float* bo    = (const float*)d_in[16];
    const float* fw1   = (const float*)d_in[17];
    const float* fb1   = (const float*)d_in[18];
    const float* fw2   = (const float*)d_in[19];
    const float* fb2   = (const float*)d_in[20];
    float* out = (float*)d_out;

    // workspace layout (floats); every region is fully written before read.
    float* ws    = (float*)d_ws;
    float* part3 = ws;                              // 2048*128
    float* fin3  = part3 + (size_t)2048 * 128;      // 64
    float* y2    = fin3 + 64;                       // 1,048,576
    float* part2 = y2 + (size_t)1048576;            // 2048*128
    float* fin2  = part2 + (size_t)2048 * 128;      // 128
    float* pwb   = fin2 + 128;                      // 2,097,152
    float* redb  = pwb + (size_t)2097152;           // 2,097,152
    float* tokb  = redb + (size_t)2097152;          // 2,099,200
    float* kb    = tokb + (size_t)2099200;          // 2,099,200
    float* vb    = kb + (size_t)2099200;            // 2,099,200

    k_conv_stats<<<dim3(2048), dim3(32), 0, stream>>>(x, w3d1, w3d2, part3);
    k_fin3<<<dim3(1), dim3(32), 0, stream>>>(part3, fin3);
    k_conv_apply<<<dim3(2048), dim3(32), 0, stream>>>(x, w3d1, w3d2, bn3g, bn3b,
                                                      fin3, y2);
    k_inv_proj<<<dim3(2048), dim3(32), 0, stream>>>(y2, w_dw, w_pw, w_red,
                                                    pwb, redb, part2);
    k_fin2<<<dim3(1), dim3(64), 0, stream>>>(part2, fin2);
    k_involution_tok<<<dim3(1024), dim3(32), 0, stream>>>(pwb, redb, fin2, bn2g,
                                                          bn2b, w_spn, cls, tokb);
    k_kv_proj<<<dim3(2050), dim3(32), 0, stream>>>(tokb, wk, wv, kb, vb);
    k_attn_ffn<<<dim3(32), dim3(256), 0, stream>>>(tokb, kb, vb, wq, wo, bo,
                                                   fw1, fb1, fw2, fb2, out);
}